// SPNet_83846351552871
// MI455X (gfx1250) — compile-verified
//
#include <hip/hip_runtime.h>

typedef __attribute__((ext_vector_type(2))) float v2f;
typedef __attribute__((ext_vector_type(4))) float v4f;
typedef __attribute__((ext_vector_type(8))) float v8f;

__device__ __forceinline__ int iclamp(int v, int lo, int hi) {
  return v < lo ? lo : (v > hi ? hi : v);
}

// ---------------------------------------------------------------------------
// 3x3 stride-2 pad-1 conv as implicit GEMM on V_WMMA_F32_16X16X4_F32.
//   M = OC (padded to 16), K = IC*9 (padded to multiple of 4), N = OH*OW.
// One 16x16 output tile (16 pixels x 16 channels) per wave per iteration.
// A-fragment (16x4 f32): lane&15 = M(row), lane>>4 selects K pair {0,1}/{2,3}.
// B-fragment (4x16 f32): lane&15 = N(col), same K-pair striping.
// C/D (16x16 f32, 8 VGPRs): elem(M = v + 8*(lane>>4), N = lane&15).
// All out-of-range gathers are done branch-free: addresses are clamped into
// the buffer and the value is cndmask'ed to zero, so EXEC never changes and
// the WMMA loop stays a clean load/select/wmma stream.
// Leaky ReLU fused at store. Output written compact [OC, OH*OW] to LDS.
// ---------------------------------------------------------------------------
template<bool IN_GLOBAL, int IC, int IH, int IW, int OC, int OH, int OW>
__device__ __forceinline__ void conv3x3s2_wmma(
    const float* __restrict__ gin,   // global input  (IN_GLOBAL = true)
    const float*              lin,   // LDS input     (IN_GLOBAL = false)
    float*                    lout,  // LDS output [OC, OH*OW]
    const float* __restrict__ w,     // OIHW weights, flat [OC, IC*9]
    const float* __restrict__ bias)  // [OC]
{
  constexpr int K      = IC * 9;
  constexpr int nsteps = (K + 3) >> 2;
  constexpr int N      = OH * OW;      // multiple of 16 for all layers here
  constexpr int ntiles = N >> 4;
  const int lane   = threadIdx.x & 31;
  const int wave   = threadIdx.x >> 5;
  const int nwaves = blockDim.x >> 5;
  const int mrow   = lane & 15;        // M for A, N(col) for B
  const int khalf  = (lane >> 4) << 1; // 0 or 2: which K pair this lane holds
  const int arow   = mrow < OC ? mrow : 0;  // clamped weight row

  for (int t = wave; t < ntiles; t += nwaves) {
    const int n  = (t << 4) + mrow;    // output pixel index for B columns
    const int oy = n / OW;             // OW is a power of two -> shift
    const int ox = n - oy * OW;
    v8f acc = {0.f, 0.f, 0.f, 0.f, 0.f, 0.f, 0.f, 0.f};

#pragma unroll
    for (int s = 0; s < nsteps; ++s) {
      const int k0 = (s << 2) + khalf;
      v2f a, bf;
#pragma unroll
      for (int j = 0; j < 2; ++j) {
        const int kk = k0 + j;
        const int kc = kk < K ? kk : K - 1;
        // ---- A: weight element (branch-free, clamped address + mask) ----
        const float wraw = w[arow * K + kc];
        const float wval = (mrow < OC && kk < K) ? wraw : 0.f;
        // ---- B: im2col patch element -----------------------------------
        const int ic = kc / 9;
        const int r  = kc - ic * 9;
        const int ky = r / 3;
        const int kx = r - ky * 3;
        const int iy = 2 * oy - 1 + ky;
        const int ix = 2 * ox - 1 + kx;
        const bool inb = (kk < K) && (iy >= 0) && (iy < IH) &&
                         (ix >= 0) && (ix < IW);
        const int off = (ic * IH + iclamp(iy, 0, IH - 1)) * IW +
                        iclamp(ix, 0, IW - 1);
        const float raw = IN_GLOBAL ? gin[off] : lin[off];
        const float bval = inb ? raw : 0.f;
        if (j == 0) { a.x = wval; bf.x = bval; }
        else        { a.y = wval; bf.y = bval; }
      }
      acc = __builtin_amdgcn_wmma_f32_16x16x4_f32(
                /*neg_a=*/false, a, /*neg_b=*/false, bf,
                /*c_mod=*/(short)0, acc, /*reuse_a=*/false, /*reuse_b=*/false);
    }

    // Store D: lane L, vgpr v -> (M = v + 8*(L>>4), N = t*16 + (L&15))
#pragma unroll
    for (int v = 0; v < 8; ++v) {
      const int oc = v + ((lane >> 4) << 3);
      if (oc < OC) {
        float val = acc[v] + bias[oc];
        lout[oc * N + n] = (val >= 0.f) ? val : 0.01f * val;   // leaky relu
      }
    }
  }
}

// ---------------------------------------------------------------------------
// Palette branch: one workgroup per batch image, 32 waves. Only the receptive
// field of the final 4x4 corner is computed:
//   x[:, :128,:128] -> h1[6,64,64] -> h2[12,32,32] -> h3[12,16,16]
//   -> h4[12,8,8] -> h5[6,4,4] -> pal[3,16]
// All intermediates live in (dynamic) LDS: 40800 floats = 159.4 KB.
// ---------------------------------------------------------------------------
__global__ void __launch_bounds__(1024)
palette_kernel(const float* __restrict__ x,
               const float* __restrict__ w1, const float* __restrict__ b1,
               const float* __restrict__ w2, const float* __restrict__ b2,
               const float* __restrict__ w3, const float* __restrict__ b3,
               const float* __restrict__ w4, const float* __restrict__ b4,
               const float* __restrict__ w5, const float* __restrict__ b5,
               const float* __restrict__ w6, const float* __restrict__ b6,
               float* __restrict__ pal_out)          // [16, 3, 16]
{
  extern __shared__ float smem[];
  float* h1 = smem;                     //  6*64*64 = 24576
  float* h2 = h1 + 6 * 64 * 64;         // 12*32*32 = 12288
  float* h3 = h2 + 12 * 32 * 32;        // 12*16*16 =  3072
  float* h4 = h3 + 12 * 16 * 16;        // 12* 8* 8 =   768
  float* h5 = h4 + 12 * 8 * 8;          //  6* 4* 4 =    96

  const int b = blockIdx.x;
  const float* xb = x + (size_t)b * 3 * 400 * 400;

  conv3x3s2_wmma<true,  3, 400, 400,  6, 64, 64>(xb,      nullptr, h1, w1, b1);
  __syncthreads();
  conv3x3s2_wmma<false, 6,  64,  64, 12, 32, 32>(nullptr, h1,      h2, w2, b2);
  __syncthreads();
  conv3x3s2_wmma<false, 12, 32,  32, 12, 16, 16>(nullptr, h2,      h3, w3, b3);
  __syncthreads();
  conv3x3s2_wmma<false, 12, 16,  16, 12,  8,  8>(nullptr, h3,      h4, w4, b4);
  __syncthreads();
  conv3x3s2_wmma<false, 12,  8,   8,  6,  4,  4>(nullptr, h4,      h5, w5, b5);
  __syncthreads();

  // conv6: 1x1, relu — only the 48 palette values. k = row*4+col matches the
  // h5 pixel index n = oy*4+ox directly.
  if (threadIdx.x < 48) {
    const int c = threadIdx.x >> 4;
    const int k = threadIdx.x & 15;
    float s = b6[c];
#pragma unroll
    for (int ic = 0; ic < 6; ++ic) s += w6[c * 6 + ic] * h5[ic * 16 + k];
    pal_out[b * 48 + c * 16 + k] = (s > 0.f) ? s : 0.f;
  }
}

// ---------------------------------------------------------------------------
// Main branch. Softmax is strictly monotone per pixel, so
// (probs == max probs) <=> (logits == max logits): skip the exp entirely.
// Each thread owns 4 consecutive pixels (b128 non-temporal loads/stores),
// reads the 16 class planes once (163.8 MB total, streamed with TH=NT), and
// keeps tie semantics by summing pal[:,k] over every k attaining the max.
// HBM roofline: (163.8 + 30.7) MB / 23.3 TB/s ~= 8.4 us.
// ---------------------------------------------------------------------------
__global__ void __launch_bounds__(256)
gather_kernel(const float* __restrict__ logits,   // [16,16,400,400]
              const float* __restrict__ pal_ws,   // [16,3,16]
              float* __restrict__ out)            // [16,3,400,400]
{
  constexpr int HW = 400 * 400;
  __shared__ float pal[48];
  const int b = blockIdx.y;
  if (threadIdx.x < 48) pal[threadIdx.x] = pal_ws[b * 48 + threadIdx.x];
  __syncthreads();

  const int pix = (blockIdx.x * 256 + threadIdx.x) * 4;
  if (pix >= HW) return;

  const float* base = logits + (size_t)b * 16 * HW + pix;

  float v[16][4];
#pragma unroll
  for (int k = 0; k < 16; ++k) {
    const v4f t = __builtin_nontemporal_load(
        reinterpret_cast<const v4f*>(base + (size_t)k * HW));
    v[k][0] = t.x; v[k][1] = t.y; v[k][2] = t.z; v[k][3] = t.w;
  }

  float mx[4] = {v[0][0], v[0][1], v[0][2], v[0][3]};
#pragma unroll
  for (int k = 1; k < 16; ++k) {
#pragma unroll
    for (int c = 0; c < 4; ++c) mx[c] = fmaxf(mx[c], v[k][c]);
  }

  float acc0[4] = {0.f, 0.f, 0.f, 0.f};
  float acc1[4] = {0.f, 0.f, 0.f, 0.f};
  float acc2[4] = {0.f, 0.f, 0.f, 0.f};
#pragma unroll
  for (int k = 0; k < 16; ++k) {
    const float p0 = pal[k], p1 = pal[16 + k], p2 = pal[32 + k];
#pragma unroll
    for (int c = 0; c < 4; ++c) {
      const float m = (v[k][c] == mx[c]) ? 1.f : 0.f;
      acc0[c] = fmaf(m, p0, acc0[c]);
      acc1[c] = fmaf(m, p1, acc1[c]);
      acc2[c] = fmaf(m, p2, acc2[c]);
    }
  }

  float* o = out + (size_t)b * 3 * HW + pix;
  const v4f o0 = {acc0[0], acc0[1], acc0[2], acc0[3]};
  const v4f o1 = {acc1[0], acc1[1], acc1[2], acc1[3]};
  const v4f o2 = {acc2[0], acc2[1], acc2[2], acc2[3]};
  __builtin_nontemporal_store(o0, reinterpret_cast<v4f*>(o));
  __builtin_nontemporal_store(o1, reinterpret_cast<v4f*>(o + HW));
  __builtin_nontemporal_store(o2, reinterpret_cast<v4f*>(o + 2 * HW));
}

// ---------------------------------------------------------------------------
extern "C" void kernel_launch(void* const* d_in, const int* in_sizes, int n_in,
                              void* d_out, int out_size, void* d_ws, size_t ws_size,
                              hipStream_t stream)
{
  const float* x    = (const float*)d_in[0];
  const float* blog = (const float*)d_in[1];
  const float* w1   = (const float*)d_in[2];   const float* b1 = (const float*)d_in[3];
  const float* w2   = (const float*)d_in[4];   const float* b2 = (const float*)d_in[5];
  const float* w3   = (const float*)d_in[6];   const float* b3 = (const float*)d_in[7];
  const float* w4   = (const float*)d_in[8];   const float* b4 = (const float*)d_in[9];
  const float* w5   = (const float*)d_in[10];  const float* b5 = (const float*)d_in[11];
  const float* w6   = (const float*)d_in[12];  const float* b6 = (const float*)d_in[13];

  float* pal = (float*)d_ws;   // 16*3*16 = 768 floats of scratch

  const size_t lds_bytes =
      (size_t)(6 * 64 * 64 + 12 * 32 * 32 + 12 * 16 * 16 + 12 * 8 * 8 + 6 * 4 * 4)
      * sizeof(float);  // 163200 B, < 320 KB/WGP

  palette_kernel<<<16, 1024, lds_bytes, stream>>>(
      x, w1, b1, w2, b2, w3, b3, w4, b4, w5, b5, w6, b6, pal);

  constexpr int HW = 400 * 400;
  const int blocks_x = (HW / 4 + 255) / 256;   // 157
  gather_kernel<<<dim3(blocks_x, 16), 256, 0, stream>>>(
      blog, pal, (float*)d_out);
}